// BuddyPool_54468775248442
// MI455X (gfx1250) — compile-verified
//
#include <hip/hip_runtime.h>

typedef __attribute__((ext_vector_type(2))) float v2f;
typedef __attribute__((ext_vector_type(8))) float v8f;
typedef unsigned int u32;
typedef __attribute__((ext_vector_type(4))) u32 v4u;
typedef __attribute__((ext_vector_type(8))) int v8i;
typedef __attribute__((ext_vector_type(4))) int v4i;

#define B_      64
#define K_      5
#define D_      1024
#define NPIX    1024   // 32*32
#define W_      32
#define TILE_PX 128
#define NTILES  8      // NPIX / TILE_PX
#define CHUNK   64
#define NCHUNK  16     // D_ / CHUNK
#define LSTR    68     // padded LDS row stride (floats); TDM pad: 4 DW every 64 DW

// ---------------------------------------------------------------------------
// Kernel 1: normalize cue rows. One block per (b,k) row of 1024 floats.
// ---------------------------------------------------------------------------
__global__ void __launch_bounds__(256) cue_norm_kernel(const float* __restrict__ cue,
                                                       float* __restrict__ cueN) {
    const int row = blockIdx.x;
    const int tid = threadIdx.x;
    const float* src = cue + (size_t)row * D_;
    float* dst = cueN + (size_t)row * D_;

    float4 x = ((const float4*)src)[tid];
    float ss = x.x * x.x + x.y * x.y + x.z * x.z + x.w * x.w;
    #pragma unroll
    for (int m = 16; m; m >>= 1) ss += __shfl_xor(ss, m, 32);

    __shared__ float red[8];
    __shared__ float s_scale;
    if ((tid & 31) == 0) red[tid >> 5] = ss;
    __syncthreads();
    if (tid == 0) {
        float s = 0.f;
        #pragma unroll
        for (int i = 0; i < 8; ++i) s += red[i];
        s_scale = 1.f / fmaxf(sqrtf(s), 1e-12f);
    }
    __syncthreads();
    const float sc = s_scale;
    x.x *= sc; x.y *= sc; x.z *= sc; x.w *= sc;
    ((float4*)dst)[tid] = x;
}

// ---------------------------------------------------------------------------
// Kernel 2: per-batch GEMM sims = cueN * patches^T via v_wmma_f32_16x16x4_f32.
// Patch tiles are streamed by the Tensor Data Mover (tensor_load_to_lds) with
// hardware LDS padding (stride 68 floats), double-buffered and overlapped with
// WMMA compute. Per-pixel squared norms fall out of the B-fragments for free.
// Grid: (b * NTILES + tile), block: 256 threads = 8 waves.
// ---------------------------------------------------------------------------
__device__ __forceinline__ void tdm_load_patch_tile(const float* gsrc, u32 lds_off) {
    const unsigned long long ga = (unsigned long long)(uintptr_t)gsrc; // byte address
    v4u g0;
    g0[0] = 1u;                                   // count=1, user mode
    g0[1] = lds_off;                              // lds_addr (bytes)
    g0[2] = (u32)ga;                              // global_addr[31:0]
    g0[3] = (u32)(ga >> 32) | 0x80000000u;        // global_addr[56:32] | type=2
    v8i g1;
    g1[0] = (int)((2u << 16)                      // data_size = 4 bytes
                | (1u << 20)                      // pad_enable
                | (5u << 22)                      // pad_interval: 64 DWORDs
                | (3u << 25));                    // pad_amount: 4 DWORDs
    g1[1] = (int)(CHUNK << 16);                   // tensor_dim0 = 64 (bits 79:48 lo)
    g1[2] = (int)(TILE_PX << 16);                 // tensor_dim1 = 128 (bits 111:80 lo)
    g1[3] = (int)(CHUNK << 16);                   // tile_dim0 = 64 (bits 127:112)
    g1[4] = TILE_PX;                              // tile_dim1 = 128; tile_dim2 = 0
    g1[5] = D_;                                   // tensor_dim0_stride = 1024
    g1[6] = 0;
    g1[7] = 0;
    v4i gz4 = {0, 0, 0, 0};                       // 2-D tile: groups 2/3 unused
    v8i gz8 = {0, 0, 0, 0, 0, 0, 0, 0};
    __builtin_amdgcn_tensor_load_to_lds(g0, g1, gz4, gz4, gz8, 0);
}

__global__ void __launch_bounds__(256) sims_argmax_kernel(const float* __restrict__ cueN,
                                                          const float* __restrict__ patches,
                                                          float* __restrict__ pvals,
                                                          int* __restrict__ pidx) {
    const int b    = blockIdx.x / NTILES;
    const int tile = blockIdx.x % NTILES;
    const int tid  = threadIdx.x;
    const int wave = tid >> 5;
    const int lane = tid & 31;
    const int half = lane >> 4;      // K-pair selector for A/B frags
    const int mrow = lane & 15;      // M (cue) for A, N (pixel) for B/C

    __shared__ float lds_patch[2 * TILE_PX * LSTR]; // 69.6 KB, double-buffered
    __shared__ float lds_cue[2 * 16 * LSTR];        //  8.7 KB, double-buffered
    __shared__ float red_val[8 * K_];
    __shared__ int   red_idx[8 * K_];

    const float* patchBase = patches + ((size_t)b * NPIX + (size_t)tile * TILE_PX) * D_;
    const float* cueBase   = cueN + (size_t)b * K_ * D_;

    // ---- prologue: kick off TDM for chunk 0, stage cue chunk 0 ----
    if (wave == 0)
        tdm_load_patch_tile(patchBase, (u32)(uintptr_t)&lds_patch[0]);
    for (int i = tid; i < 16 * CHUNK; i += 256) {
        const int r = i >> 6, c = i & 63;
        lds_cue[r * LSTR + c] = (r < K_) ? cueBase[(size_t)r * D_ + c] : 0.f;
    }

    v8f acc = {0.f, 0.f, 0.f, 0.f, 0.f, 0.f, 0.f, 0.f};
    float nacc = 0.f;                // partial |patch|^2 (this lane's d-subset)
    const int prow = wave * 16 + mrow;

    for (int ch = 0; ch < NCHUNK; ++ch) {
        const int cur = ch & 1;
        // ---- prefetch chunk ch+1 into the other buffer ----
        if (ch < NCHUNK - 1) {
            const int nxt = cur ^ 1;
            if (wave == 0)
                tdm_load_patch_tile(patchBase + (ch + 1) * CHUNK,
                                    (u32)(uintptr_t)&lds_patch[nxt * TILE_PX * LSTR]);
            const int dB = (ch + 1) * CHUNK;
            for (int i = tid; i < 16 * CHUNK; i += 256) {
                const int r = i >> 6, c = i & 63;
                lds_cue[(nxt * 16 + r) * LSTR + c] =
                    (r < K_) ? cueBase[(size_t)r * D_ + dB + c] : 0.f;
            }
        }
        // ---- ensure chunk ch has landed (TDM in-order per wave) ----
        if (wave == 0) {
            if (ch < NCHUNK - 1) __builtin_amdgcn_s_wait_tensorcnt(1);
            else                 __builtin_amdgcn_s_wait_tensorcnt(0);
        }
        __syncthreads();   // barrier A: buffer `cur` ready & visible

        const float* pb = &lds_patch[cur * TILE_PX * LSTR];
        const float* cb = &lds_cue[cur * 16 * LSTR];
        #pragma unroll
        for (int kk = 0; kk < CHUNK / 4; ++kk) {
            const int col = kk * 4 + half * 2;
            // A frag (ISA 7.12.2): M=lane&15; lanes 0-15 K={0,1}, 16-31 K={2,3}
            v2f a = *(const v2f*)(&cb[mrow * LSTR + col]);
            // B frag (mirrored): N=lane&15, half-wave selects K pair
            v2f bb = *(const v2f*)(&pb[prow * LSTR + col]);
            nacc = fmaf(bb.x, bb.x, nacc);
            nacc = fmaf(bb.y, bb.y, nacc);
            acc = __builtin_amdgcn_wmma_f32_16x16x4_f32(false, a, false, bb,
                                                        (short)0, acc, false, false);
        }
        __syncthreads();   // barrier B: everyone done reading buffer `cur`
    }

    // ---- per-pixel norm: lane L and L+16 hold complementary d-subsets ----
    const float norm2 = nacc + __shfl_xor(nacc, 16, 32);
    const float scale = 1.f / fmaxf(sqrtf(norm2), 1e-12f);
    const int myPix   = tile * TILE_PX + prow;

    // C layout: lanes 0-15 => N=lane, acc[r] = M=r (cues 0..4 live here).
    #pragma unroll
    for (int k = 0; k < K_; ++k) {
        float bv = acc[k] * scale;
        int   bi = myPix;
        #pragma unroll
        for (int m = 1; m < 16; m <<= 1) {
            const float ov = __shfl_xor(bv, m, 16);
            const int   oi = __shfl_xor(bi, m, 16);
            if (ov > bv || (ov == bv && oi < bi)) { bv = ov; bi = oi; }
        }
        if (lane == 0) { red_val[wave * K_ + k] = bv; red_idx[wave * K_ + k] = bi; }
    }
    __syncthreads();

    if (tid < K_) {
        float bv = red_val[tid];
        int   bi = red_idx[tid];
        for (int w = 1; w < 8; ++w) {
            const float ov = red_val[w * K_ + tid];
            const int   oi = red_idx[w * K_ + tid];
            if (ov > bv || (ov == bv && oi < bi)) { bv = ov; bi = oi; }
        }
        const int o = (b * NTILES + tile) * K_ + tid;
        pvals[o] = bv;
        pidx[o]  = bi;
    }
}

// ---------------------------------------------------------------------------
// Kernel 3: reduce tile partials -> final argmax; 3x3 zero-padded neighborhood
// mean from the ORIGINAL patches. One block per (b,k), 256 threads x float4.
// ---------------------------------------------------------------------------
__global__ void __launch_bounds__(256) gather_kernel(const float* __restrict__ patches,
                                                     const float* __restrict__ pvals,
                                                     const int* __restrict__ pidx,
                                                     float* __restrict__ out) {
    const int b = blockIdx.x / K_;
    const int k = blockIdx.x % K_;

    __shared__ int s_idx;
    if (threadIdx.x == 0) {
        float bv = -3.402823466e38f;
        int   bi = 0x7fffffff;
        for (int t = 0; t < NTILES; ++t) {
            const int o = (b * NTILES + t) * K_ + k;
            const float v = pvals[o];
            const int   i = pidx[o];
            if (v > bv || (v == bv && i < bi)) { bv = v; bi = i; }
        }
        s_idx = bi;
    }
    __syncthreads();

    const int idx = s_idx;
    const int cy = idx >> 5;
    const int cx = idx & (W_ - 1);

    const float* pb = patches + (size_t)b * NPIX * D_;
    float4 s = {0.f, 0.f, 0.f, 0.f};
    #pragma unroll
    for (int dy = -1; dy <= 1; ++dy) {
        const int y = cy + dy;
        if ((unsigned)y >= (unsigned)W_) continue;
        #pragma unroll
        for (int dx = -1; dx <= 1; ++dx) {
            const int x = cx + dx;
            if ((unsigned)x >= (unsigned)W_) continue;
            const float4 v = *(const float4*)(pb + (size_t)(y * W_ + x) * D_ + threadIdx.x * 4);
            s.x += v.x; s.y += v.y; s.z += v.z; s.w += v.w;
        }
    }
    const float inv9 = 1.f / 9.f;
    s.x *= inv9; s.y *= inv9; s.z *= inv9; s.w *= inv9;
    *(float4*)(out + (size_t)(b * K_ + k) * D_ + threadIdx.x * 4) = s;
}

// ---------------------------------------------------------------------------
extern "C" void kernel_launch(void* const* d_in, const int* in_sizes, int n_in,
                              void* d_out, int out_size, void* d_ws, size_t ws_size,
                              hipStream_t stream) {
    const float* cue     = (const float*)d_in[0];   // (64,5,1024) f32
    const float* patches = (const float*)d_in[1];   // (64,32,32,1024) f32
    float* out = (float*)d_out;                     // (64,5,1024) f32

    float* cueN  = (float*)d_ws;                          // 320*1024 floats
    float* pvals = cueN + (size_t)B_ * K_ * D_;           // 64*8*5 floats
    int*   pidx  = (int*)(pvals + (size_t)B_ * NTILES * K_);

    cue_norm_kernel<<<B_ * K_, 256, 0, stream>>>(cue, cueN);
    sims_argmax_kernel<<<B_ * NTILES, 256, 0, stream>>>(cueN, patches, pvals, pidx);
    gather_kernel<<<B_ * K_, 256, 0, stream>>>(patches, pvals, pidx, out);
}